// TorchConfusionMatrixMetric_36034775613922
// MI455X (gfx1250) — compile-verified
//
#include <hip/hip_runtime.h>

#define NCLS 32
#define BINS (NCLS * NCLS)   // 1024 bins

typedef float v2f __attribute__((ext_vector_type(2)));
typedef float v8f __attribute__((ext_vector_type(8)));

// ---------------------------------------------------------------------------
// Phase 1: per-workgroup privatized histogram in LDS (ds_add_u32), vectorized
// b128 global loads. Memory-bound: 64 MB @ 23.3 TB/s ~ 2.8 us floor.
// Writes one 1024-bin u32 partial histogram per workgroup into d_ws.
// ---------------------------------------------------------------------------
__global__ void cm_partial_hist(const int* __restrict__ yt,
                                const int* __restrict__ yp,
                                unsigned* __restrict__ ws, int n) {
  __shared__ unsigned h[BINS];
  for (int i = threadIdx.x; i < BINS; i += blockDim.x) h[i] = 0u;
  __syncthreads();

  const int tid = blockIdx.x * blockDim.x + threadIdx.x;
  const int nth = gridDim.x * blockDim.x;
  const int n4  = n >> 2;
  const int4* yt4 = (const int4*)yt;
  const int4* yp4 = (const int4*)yp;

  for (int i = tid; i < n4; i += nth) {
    int4 t = yt4[i];           // global_load_b128
    int4 p = yp4[i];           // global_load_b128
    if (t.x) atomicAdd(&h[((t.x & 31) << 5) | (p.x & 31)], 1u);
    if (t.y) atomicAdd(&h[((t.y & 31) << 5) | (p.y & 31)], 1u);
    if (t.z) atomicAdd(&h[((t.z & 31) << 5) | (p.z & 31)], 1u);
    if (t.w) atomicAdd(&h[((t.w & 31) << 5) | (p.w & 31)], 1u);
  }
  // tail (n % 4 elements)
  const int base = n4 << 2;
  if (tid < n - base) {
    int t = yt[base + tid], p = yp[base + tid];
    if (t) atomicAdd(&h[((t & 31) << 5) | (p & 31)], 1u);
  }
  __syncthreads();

  unsigned* dst = ws + (size_t)blockIdx.x * BINS;
  for (int i = threadIdx.x; i < BINS; i += blockDim.x) dst[i] = h[i];
}

// ---------------------------------------------------------------------------
// Phase 2: out(1x1024) = ones(1xG) @ partials(Gx1024) via V_WMMA_F32_16X16X4_F32.
// One wave per 16-column tile (64 waves). A = all-ones 16x4 (every D row gets
// the column sum); B = 4x16 tile of partials, K-steps of 4 over G.
// Counts < 2^24, so f32 accumulation is exact and deterministic.
// 32-bit B 4x16 layout: lanes 0-15 hold N=0..15 with K=0 (v0) / K=1 (v1);
// lanes 16-31 hold K=2 (v0) / K=3 (v1).
// ---------------------------------------------------------------------------
__global__ void cm_wmma_reduce(const unsigned* __restrict__ ws,
                               float* __restrict__ out, int G) {
  const int lane = threadIdx.x;                 // 0..31, wave32
  const int col  = blockIdx.x * 16 + (lane & 15);
  const int krow = (lane >> 4) << 1;            // 0 for lanes 0-15, 2 for 16-31

  v8f acc = {};
  v2f a;
  a.x = 1.0f;
  a.y = 1.0f;

  for (int g = 0; g < G; g += 4) {
    v2f b;
    b.x = (float)ws[(size_t)(g + krow)     * BINS + col];
    b.y = (float)ws[(size_t)(g + krow + 1) * BINS + col];
    acc = __builtin_amdgcn_wmma_f32_16x16x4_f32(
        /*neg_a=*/false, a, /*neg_b=*/false, b,
        /*c_mod=*/(short)0, acc, /*reuse_a=*/false, /*reuse_b=*/false);
  }

  // D row M=0 lives in acc[0] for lanes 0-15 (N = lane).
  if (lane < 16) out[col] = acc[0];
}

// ---------------------------------------------------------------------------
// Fallback path (only if ws_size is pathologically small): zero output, then
// LDS histogram with float global atomics straight into d_out.
// ---------------------------------------------------------------------------
__global__ void cm_zero(float* __restrict__ out) {
  out[blockIdx.x * blockDim.x + threadIdx.x] = 0.0f;
}

__global__ void cm_direct(const int* __restrict__ yt,
                          const int* __restrict__ yp,
                          float* __restrict__ out, int n) {
  __shared__ unsigned h[BINS];
  for (int i = threadIdx.x; i < BINS; i += blockDim.x) h[i] = 0u;
  __syncthreads();

  const int tid = blockIdx.x * blockDim.x + threadIdx.x;
  const int nth = gridDim.x * blockDim.x;
  const int n4  = n >> 2;
  const int4* yt4 = (const int4*)yt;
  const int4* yp4 = (const int4*)yp;
  for (int i = tid; i < n4; i += nth) {
    int4 t = yt4[i];
    int4 p = yp4[i];
    if (t.x) atomicAdd(&h[((t.x & 31) << 5) | (p.x & 31)], 1u);
    if (t.y) atomicAdd(&h[((t.y & 31) << 5) | (p.y & 31)], 1u);
    if (t.z) atomicAdd(&h[((t.z & 31) << 5) | (p.z & 31)], 1u);
    if (t.w) atomicAdd(&h[((t.w & 31) << 5) | (p.w & 31)], 1u);
  }
  const int base = n4 << 2;
  if (tid < n - base) {
    int t = yt[base + tid], p = yp[base + tid];
    if (t) atomicAdd(&h[((t & 31) << 5) | (p & 31)], 1u);
  }
  __syncthreads();
  for (int i = threadIdx.x; i < BINS; i += blockDim.x) {
    unsigned v = h[i];
    if (v) atomicAdd(&out[i], (float)v);
  }
}

extern "C" void kernel_launch(void* const* d_in, const int* in_sizes, int n_in,
                              void* d_out, int out_size, void* d_ws, size_t ws_size,
                              hipStream_t stream) {
  const int* yt = (const int*)d_in[0];
  const int* yp = (const int*)d_in[1];
  float* out    = (float*)d_out;
  const int n   = in_sizes[0];

  // One 1024-bin u32 partial per workgroup in the workspace.
  size_t gmax = ws_size / (BINS * sizeof(unsigned));
  int G = (int)(gmax < 1024 ? gmax : 1024);
  G &= ~3;  // K-loop steps of 4 in the WMMA reduction

  if (G >= 4) {
    cm_partial_hist<<<G, 256, 0, stream>>>(yt, yp, (unsigned*)d_ws, n);
    cm_wmma_reduce<<<BINS / 16, 32, 0, stream>>>((const unsigned*)d_ws, out, G);
  } else {
    cm_zero<<<BINS / 256, 256, 0, stream>>>(out);
    cm_direct<<<256, 256, 0, stream>>>(yt, yp, out, n);
  }
}